// TrackSSM_26139170964334
// MI455X (gfx1250) — compile-verified
//
#include <hip/hip_runtime.h>
#include <hip/hip_bf16.h>

#define BATCH 256
#define SEQ   100
#define DM    256
#define DI    512
#define DID   512
#define NST   16
#define DTR   16
#define DCONV 4
#define NLAYERS 3

typedef float v2f __attribute__((ext_vector_type(2)));
typedef float v8f __attribute__((ext_vector_type(8)));

__device__ __forceinline__ float silu_f(float x) {
    return x / (1.0f + __expf(-x));
}
__device__ __forceinline__ float softplus_f(float x) {
    return (x > 20.0f) ? x : log1pf(__expf(x));
}

// ---------------------------------------------------------------------------
// K0: fold embed into m_in_w:  Wc[8][1024] = W_embed @ m_in_w ; bc = b_embed @ m_in_w
// ---------------------------------------------------------------------------
__global__ void prep_wc_kernel(const float* __restrict__ W_embed,
                               const float* __restrict__ b_embed,
                               const float* __restrict__ m_in_w,
                               float* __restrict__ Wc, float* __restrict__ bc) {
    int j = blockIdx.x * blockDim.x + threadIdx.x;   // 0..1023
    if (j >= 2 * DI) return;
    float acc[8] = {0.f, 0.f, 0.f, 0.f, 0.f, 0.f, 0.f, 0.f};
    float accb = 0.f;
    for (int i = 0; i < DM; ++i) {
        float mv = m_in_w[(size_t)i * (2 * DI) + j];
#pragma unroll
        for (int k = 0; k < 8; ++k) acc[k] += W_embed[k * DM + i] * mv;
        accb += b_embed[i] * mv;
    }
#pragma unroll
    for (int k = 0; k < 8; ++k) Wc[k * (2 * DI) + j] = acc[k];
    bc[j] = accb;
}

// ---------------------------------------------------------------------------
// Generic fp32 WMMA GEMM: C[M,N] = act(A[M,K](lda) @ B[K,N](ldb) + bias)
// One wave (32 threads) computes one 16x16 tile via V_WMMA_F32_16X16X4_F32.
// M % 16 == 0, N % 16 == 0, K % 4 == 0. grid = (N/16, M/16), block = 32.
// ACT: 0 = none, 1 = softplus
// ---------------------------------------------------------------------------
template <int ACT>
__global__ void wmma_gemm_f32(const float* __restrict__ A, int lda,
                              const float* __restrict__ B, int ldb,
                              const float* __restrict__ bias,
                              float* __restrict__ C, int ldc, int K) {
    const int lane = threadIdx.x;        // 0..31
    const int half = lane >> 4;          // 0 | 1
    const int l16  = lane & 15;
    const int m0 = blockIdx.y * 16;
    const int n0 = blockIdx.x * 16;

    const float* Arow = A + (size_t)(m0 + l16) * lda + 2 * half;   // A[m][k + 2*half .. +1]
    const float* Bcol = B + (size_t)(2 * half) * ldb + n0 + l16;   // B[k + 2*half][n]

    v8f acc = {};
    for (int k = 0; k < K; k += 4) {
        v2f a, b;
        a.x = Arow[k];
        a.y = Arow[k + 1];
        b.x = Bcol[(size_t)k * ldb];
        b.y = Bcol[(size_t)(k + 1) * ldb];
        acc = __builtin_amdgcn_wmma_f32_16x16x4_f32(
            /*neg_a=*/false, a, /*neg_b=*/false, b,
            /*c_mod=*/(short)0, acc, /*reuse_a=*/false, /*reuse_b=*/false);
    }

    const int col = n0 + l16;
    const float bv = bias ? bias[col] : 0.0f;
#pragma unroll
    for (int v = 0; v < 8; ++v) {
        int row = m0 + v + half * 8;
        float val = acc[v] + bv;
        if (ACT == 1) val = softplus_f(val);
        C[(size_t)row * ldc + col] = val;
    }
}

// ---------------------------------------------------------------------------
// K3: depthwise causal conv (DCONV=4) + SiLU:  xc[row,d]
// ---------------------------------------------------------------------------
__global__ void conv_silu_kernel(const float* __restrict__ xs,
                                 const float* __restrict__ convw,  // (512,4)
                                 const float* __restrict__ convb,  // (512)
                                 float* __restrict__ xc) {
    int row = blockIdx.x;        // b*SEQ + t
    int d = threadIdx.x;         // 0..511
    int t = row % SEQ;
    float acc = convb[d];
#pragma unroll
    for (int k = 0; k < DCONV; ++k) {
        int tt = t - (DCONV - 1) + k;
        if (tt >= 0)
            acc += xs[(size_t)(row - (DCONV - 1) + k) * DI + d] * convw[d * DCONV + k];
    }
    xc[(size_t)row * DI + d] = silu_f(acc);
}

// ---------------------------------------------------------------------------
// K6: SSM scan over SEQ steps. One block per batch b, one thread per d.
// Only the final-step y is needed:  yfin = (sum_n h99*C99 + D*x99) * silu(z99)
// ---------------------------------------------------------------------------
__global__ void scan_kernel(const float* __restrict__ dt,    // (25600,512)
                            const float* __restrict__ xc,    // (25600,512)
                            const float* __restrict__ proj,  // (25600,48)
                            const float* __restrict__ A_log, // (512,16)
                            const float* __restrict__ Dp,    // (512)
                            const float* __restrict__ z99,   // (256,512)
                            float* __restrict__ yfin) {      // (256,512)
    int b = blockIdx.x;
    int d = threadIdx.x;
    __shared__ float Bsh[NST];
    __shared__ float Csh[NST];

    float Aa[NST];
#pragma unroll
    for (int n = 0; n < NST; ++n) Aa[n] = -__expf(A_log[d * NST + n]);

    float h[NST];
#pragma unroll
    for (int n = 0; n < NST; ++n) h[n] = 0.f;

    const size_t base = (size_t)b * SEQ;
    for (int t = 0; t < SEQ; ++t) {
        size_t r = base + t;
        if (d < NST) Bsh[d] = proj[r * 48 + DTR + d];
        __syncthreads();
        float dtv = dt[r * DI + d];
        float xv  = xc[r * DI + d];
        float dx  = dtv * xv;
#pragma unroll
        for (int n = 0; n < NST; ++n)
            h[n] = h[n] * __expf(dtv * Aa[n]) + dx * Bsh[n];
        __syncthreads();
    }
    size_t r99 = base + (SEQ - 1);
    if (d < NST) Csh[d] = proj[r99 * 48 + DTR + NST + d];
    __syncthreads();

    float y = Dp[d] * xc[r99 * DI + d];
#pragma unroll
    for (int n = 0; n < NST; ++n) y += h[n] * Csh[n];
    y *= silu_f(z99[(size_t)b * DI + d]);
    yfin[(size_t)b * DI + d] = y;
}

// ---------------------------------------------------------------------------
// K8: decoder init: bbox = x[:, -1, :4], hdec = 0
// ---------------------------------------------------------------------------
__global__ void dec_init_kernel(const float* __restrict__ x,
                                float* __restrict__ bbox,
                                float* __restrict__ hdec) {
    int i = blockIdx.x * blockDim.x + threadIdx.x;
    if (i < BATCH * 4)
        bbox[i] = x[(size_t)(i >> 2) * (SEQ * 8) + (SEQ - 1) * 8 + (i & 3)];
    if (i < BATCH * DID * NST) hdec[i] = 0.f;
}

// ---------------------------------------------------------------------------
// K9: one decoder layer. grid = 256 (b), block = 512 (d).
// ---------------------------------------------------------------------------
__global__ void dec_layer_kernel(const float* __restrict__ flow,   // (256,256)
                                 const float* __restrict__ in_w,   // (4,1024)
                                 const float* __restrict__ in_b,   // (1024)
                                 const float* __restrict__ flow_w, // (256,48)
                                 const float* __restrict__ dt_w,   // (16,512)
                                 const float* __restrict__ dt_b,   // (512)
                                 const float* __restrict__ A_log,  // (512,16)
                                 const float* __restrict__ Dp,     // (512)
                                 const float* __restrict__ out_w,  // (512,4)
                                 const float* __restrict__ out_b,  // (4)
                                 float* __restrict__ bbox,         // (256,4) in/out
                                 float* __restrict__ hdec,         // (256,512,16)
                                 float* __restrict__ final_out) {  // null or (256,4)
    int b = blockIdx.x;
    int d = threadIdx.x;  // 0..511
    __shared__ float pf[48];
    __shared__ float bbx[4];
    __shared__ float red[DID];

    if (d < 4) bbx[d] = bbox[b * 4 + d];
    if (d < 48) {
        float acc = 0.f;
        const float* fr = flow + (size_t)b * DM;
        for (int k = 0; k < DM; ++k) acc += fr[k] * flow_w[k * 48 + d];
        pf[d] = acc;
    }
    __syncthreads();

    // bh = bbox @ in_w + in_b ; e = silu(first half), r = second half
    float e = in_b[d], r = in_b[DID + d];
#pragma unroll
    for (int j = 0; j < 4; ++j) {
        e += bbx[j] * in_w[j * (2 * DID) + d];
        r += bbx[j] * in_w[j * (2 * DID) + DID + d];
    }
    e = silu_f(e);

    // dtd = softplus(pf[:16] @ dt_w + dt_b)
    float dtd = dt_b[d];
#pragma unroll
    for (int k = 0; k < DTR; ++k) dtd += pf[k] * dt_w[k * DID + d];
    dtd = softplus_f(dtd);

    // h update + y
    float y = 0.f;
    size_t hb = ((size_t)b * DID + d) * NST;
#pragma unroll
    for (int n = 0; n < NST; ++n) {
        float Av = -__expf(A_log[d * NST + n]);
        float h = hdec[hb + n];
        h = h * __expf(dtd * Av) + e * (dtd * pf[DTR + n]);
        hdec[hb + n] = h;
        y += h * pf[DTR + NST + n];
    }
    y += Dp[d] * e;
    y *= silu_f(r);

    red[d] = y;
    __syncthreads();
    if (d < 4) {
        float acc = out_b[d];
        for (int k = 0; k < DID; ++k) acc += red[k] * out_w[k * 4 + d];
        bbox[b * 4 + d] = acc;
        if (final_out) final_out[b * 4 + d] = acc;
    }
}

// ---------------------------------------------------------------------------
extern "C" void kernel_launch(void* const* d_in, const int* in_sizes, int n_in,
                              void* d_out, int out_size, void* d_ws, size_t ws_size,
                              hipStream_t stream) {
    const float* x         = (const float*)d_in[0];   // (256,100,8)
    const float* W_embed   = (const float*)d_in[1];   // (8,256)
    const float* b_embed   = (const float*)d_in[2];   // (256)
    const float* m_in_w    = (const float*)d_in[3];   // (256,1024)
    const float* m_conv_w  = (const float*)d_in[4];   // (512,4)
    const float* m_conv_b  = (const float*)d_in[5];   // (512)
    const float* m_xproj_w = (const float*)d_in[6];   // (512,48)
    const float* m_dtproj_w= (const float*)d_in[7];   // (16,512)
    const float* m_dtproj_b= (const float*)d_in[8];   // (512)
    const float* m_A_log   = (const float*)d_in[9];   // (512,16)
    const float* m_D       = (const float*)d_in[10];  // (512)
    const float* m_out_w   = (const float*)d_in[11];  // (512,256)
    const float* dec_in_w  = (const float*)d_in[12];  // (3,4,1024)
    const float* dec_in_b  = (const float*)d_in[13];  // (3,1024)
    const float* dec_flow_w= (const float*)d_in[14];  // (3,256,48)
    const float* dec_dt_w  = (const float*)d_in[15];  // (3,16,512)
    const float* dec_dt_b  = (const float*)d_in[16];  // (3,512)
    const float* dec_A_log = (const float*)d_in[17];  // (3,512,16)
    const float* dec_D     = (const float*)d_in[18];  // (3,512)
    const float* dec_out_w = (const float*)d_in[19];  // (3,512,4)
    const float* dec_out_b = (const float*)d_in[20];  // (3,4)
    float* out = (float*)d_out;                       // (256,4)

    const int ROWS = BATCH * SEQ;                     // 25600

    // ---- workspace layout (256B aligned) ----
    char* ws = (char*)d_ws;
    size_t off = 0;
    auto alloc = [&](size_t bytes) {
        void* p = ws + off;
        off += (bytes + 255) & ~(size_t)255;
        return p;
    };
    float* Wc   = (float*)alloc(8 * 1024 * sizeof(float));            // (8,1024)
    float* bc   = (float*)alloc(1024 * sizeof(float));                // (1024)
    float* xs   = (float*)alloc((size_t)ROWS * DI * sizeof(float));   // (25600,512); reused for dt
    float* xc   = (float*)alloc((size_t)ROWS * DI * sizeof(float));   // (25600,512)
    float* proj = (float*)alloc((size_t)ROWS * 48 * sizeof(float));   // (25600,48)
    float* z99  = (float*)alloc((size_t)BATCH * DI * sizeof(float));  // (256,512)
    float* yfin = (float*)alloc((size_t)BATCH * DI * sizeof(float));  // (256,512)
    float* flow = (float*)alloc((size_t)BATCH * DM * sizeof(float));  // (256,256)
    float* bbox = (float*)alloc((size_t)BATCH * 4 * sizeof(float));   // (256,4)
    float* hdec = (float*)alloc((size_t)BATCH * DID * NST * sizeof(float)); // (256,512,16)
    float* dt   = xs;  // xs is dead after conv; alias

    // K0: fold embedding into input projection
    prep_wc_kernel<<<dim3(4), dim3(256), 0, stream>>>(W_embed, b_embed, m_in_w, Wc, bc);

    // K1: xs = x @ Wc[:, :512] + bc[:512]      (WMMA, M=25600, N=512, K=8)
    wmma_gemm_f32<0><<<dim3(DI / 16, ROWS / 16), dim3(32), 0, stream>>>(
        x, 8, Wc, 1024, bc, xs, DI, 8);

    // K2: z99 = x[:, -1] @ Wc[:, 512:] + bc[512:]   (WMMA, M=256, N=512, K=8)
    wmma_gemm_f32<0><<<dim3(DI / 16, BATCH / 16), dim3(32), 0, stream>>>(
        x + (SEQ - 1) * 8, SEQ * 8, Wc + DI, 1024, bc + DI, z99, DI, 8);

    // K3: xc = silu(causal_conv(xs))
    conv_silu_kernel<<<dim3(ROWS), dim3(DI), 0, stream>>>(xs, m_conv_w, m_conv_b, xc);

    // K4: proj = xc @ m_xproj_w               (WMMA, M=25600, N=48, K=512)
    wmma_gemm_f32<0><<<dim3(48 / 16, ROWS / 16), dim3(32), 0, stream>>>(
        xc, DI, m_xproj_w, 48, nullptr, proj, 48, DI);

    // K5: dt = softplus(proj[:, :16] @ m_dtproj_w + m_dtproj_b)  (WMMA, K=16, fused act)
    wmma_gemm_f32<1><<<dim3(DI / 16, ROWS / 16), dim3(32), 0, stream>>>(
        proj, 48, m_dtproj_w, DI, m_dtproj_b, dt, DI, DTR);

    // K6: SSM scan -> yfin (only final-step y is live)
    scan_kernel<<<dim3(BATCH), dim3(DI), 0, stream>>>(
        dt, xc, proj, m_A_log, m_D, z99, yfin);

    // K7: flow = yfin @ m_out_w               (WMMA, M=256, N=256, K=512)
    wmma_gemm_f32<0><<<dim3(DM / 16, BATCH / 16), dim3(32), 0, stream>>>(
        yfin, DI, m_out_w, DM, nullptr, flow, DM, DI);

    // K8: decoder state init
    dec_init_kernel<<<dim3((BATCH * DID * NST) / 256), dim3(256), 0, stream>>>(
        x, bbox, hdec);

    // K9: 3 decoder layers (sequential; tiny)
    for (int l = 0; l < NLAYERS; ++l) {
        dec_layer_kernel<<<dim3(BATCH), dim3(DID), 0, stream>>>(
            flow,
            dec_in_w  + (size_t)l * 4 * (2 * DID),
            dec_in_b  + (size_t)l * (2 * DID),
            dec_flow_w+ (size_t)l * DM * 48,
            dec_dt_w  + (size_t)l * DTR * DID,
            dec_dt_b  + (size_t)l * DID,
            dec_A_log + (size_t)l * DID * NST,
            dec_D     + (size_t)l * DID,
            dec_out_w + (size_t)l * DID * 4,
            dec_out_b + (size_t)l * 4,
            bbox, hdec,
            (l == NLAYERS - 1) ? out : nullptr);
    }
    (void)in_sizes; (void)n_in; (void)out_size; (void)ws_size;
}